// TensorProductReference_51376398795040
// MI455X (gfx1250) — compile-verified
//
#include <hip/hip_runtime.h>

// ---------------------------------------------------------------------------
// Problem constants (match the reference)
// ---------------------------------------------------------------------------
#define TP_DIM     16
#define TP_NCH     128
#define TP_NPATHS  23
#define TP_BATCH   50000
#define TP_NTILES  (TP_BATCH / 16)   // 3125, exact

typedef __attribute__((ext_vector_type(2))) float v2f;
typedef __attribute__((ext_vector_type(8))) float v8f;

// Path enumeration (l1 outer, l2 middle, l3 ascending; parity + l3<=3), 23 paths
__device__ __constant__ int PATH_L1[TP_NPATHS] =
    {0,0,0,0, 1,1,1,1,1,1, 2,2,2,2,2,2,2, 3,3,3,3,3,3};
__device__ __constant__ int PATH_L2[TP_NPATHS] =
    {0,1,2,3, 0,1,1,2,2,3, 0,1,1,2,2,3,3, 0,1,2,2,3,3};
__device__ __constant__ int PATH_L3[TP_NPATHS] =
    {0,1,2,3, 1,0,2,1,3,2, 2,1,3,0,2,1,3, 3,2,1,3,0,2};
__device__ __constant__ int IRREP_OFF[4] = {0, 1, 4, 9};

// ---------------------------------------------------------------------------
// Setup: Wigner 3j path kernels, computed on device in f64 (runs once, tiny)
// ---------------------------------------------------------------------------
__device__ double dfact(int n) {
    double r = 1.0;
    for (int i = 2; i <= n; ++i) r *= (double)i;
    return r;
}

__device__ double su2_cg(int j1, int m1, int j2, int m2, int j3, int m3) {
    if (m1 + m2 != m3) return 0.0;
    double pre = sqrt((double)(2 * j3 + 1) * dfact(j1 + j2 - j3) * dfact(j1 - j2 + j3) *
                      dfact(-j1 + j2 + j3) / dfact(j1 + j2 + j3 + 1));
    pre *= sqrt(dfact(j3 + m3) * dfact(j3 - m3) * dfact(j1 - m1) * dfact(j1 + m1) *
                dfact(j2 - m2) * dfact(j2 + m2));
    int vmin = 0;
    if (j2 - j3 - m1 > vmin) vmin = j2 - j3 - m1;
    if (j1 - j3 + m2 > vmin) vmin = j1 - j3 + m2;
    int vmax = j1 + j2 - j3;
    if (j1 - m1 < vmax) vmax = j1 - m1;
    if (j2 + m2 < vmax) vmax = j2 + m2;
    double s = 0.0;
    for (int v = vmin; v <= vmax; ++v) {
        double den = dfact(v) * dfact(j1 + j2 - j3 - v) * dfact(j1 - m1 - v) *
                     dfact(j2 + m2 - v) * dfact(j3 - j2 + m1 + v) * dfact(j3 - j1 - m2 + v);
        s += ((v & 1) ? -1.0 : 1.0) / den;
    }
    return pre * s;
}

// real->complex SH change-of-basis (e3nn convention), q = (-i)^l * q0
__device__ void build_q(int l, double re[7][7], double im[7][7]) {
    for (int i = 0; i < 7; ++i)
        for (int j = 0; j < 7; ++j) { re[i][j] = 0.0; im[i][j] = 0.0; }
    const double is2 = 0.70710678118654752440;
    for (int m = -l; m < 0; ++m) {
        re[l + m][l - m] += is2;   // q[l+m, l+|m|] = 1/sqrt2
        im[l + m][l + m] += -is2;  // q[l+m, l-|m|] = -i/sqrt2
    }
    re[l][l] = 1.0;
    for (int m = 1; m <= l; ++m) {
        double s = (m & 1) ? -1.0 : 1.0;
        re[l + m][l + m] += s * is2;  // (-1)^m/sqrt2
        im[l + m][l - m] += s * is2;  // i*(-1)^m/sqrt2
    }
    // multiply by (-i)^l
    double fr, fi;
    switch (l & 3) {
        case 0: fr = 1;  fi = 0;  break;
        case 1: fr = 0;  fi = -1; break;
        case 2: fr = -1; fi = 0;  break;
        default: fr = 0; fi = 1;  break;
    }
    for (int i = 0; i < 7; ++i)
        for (int j = 0; j < 7; ++j) {
            double a = re[i][j], b = im[i][j];
            re[i][j] = a * fr - b * fi;
            im[i][j] = a * fi + b * fr;
        }
}

// One thread per path: T[p][16][16] with the w3j block (m3 pre-summed).
// The reference's /sqrt(2l3+1) and *sqrt(2l3+1) factors cancel exactly.
__global__ void build_T_kernel(float* __restrict__ T) {
    int p = threadIdx.x;
    if (p >= TP_NPATHS) return;
    int l1 = PATH_L1[p], l2 = PATH_L2[p], l3 = PATH_L3[p];
    int d1 = 2 * l1 + 1, d2 = 2 * l2 + 1, d3 = 2 * l3 + 1;

    double Q1r[7][7], Q1i[7][7], Q2r[7][7], Q2i[7][7], Q3r[7][7], Q3i[7][7];
    build_q(l1, Q1r, Q1i);
    build_q(l2, Q2r, Q2i);
    build_q(l3, Q3r, Q3i);

    float* Tp = T + p * 256;
    for (int e = 0; e < 256; ++e) Tp[e] = 0.0f;

    // Ksub[j][l] = Re( sum_{i,k} CG(i,k) * Q1[i][j]*Q2[k][l] * sum_m conj(Q3[n][m]) )
    for (int jj = 0; jj < d1; ++jj) {
        for (int ll = 0; ll < d2; ++ll) {
            double acc = 0.0;
            for (int i = 0; i < d1; ++i) {
                for (int k = 0; k < d2; ++k) {
                    int m1 = i - l1, m2 = k - l2, m3 = m1 + m2;
                    if (m3 < -l3 || m3 > l3) continue;
                    int n = m3 + l3;
                    double cg = su2_cg(l1, m1, l2, m2, l3, m3);
                    if (cg == 0.0) continue;
                    double ar = Q1r[i][jj] * Q2r[k][ll] - Q1i[i][jj] * Q2i[k][ll];
                    double ai = Q1r[i][jj] * Q2i[k][ll] + Q1i[i][jj] * Q2r[k][ll];
                    double br = 0.0, bi = 0.0;
                    for (int mm = 0; mm < d3; ++mm) { br += Q3r[n][mm]; bi -= Q3i[n][mm]; }
                    acc += cg * (ar * br - ai * bi);
                }
            }
            Tp[(IRREP_OFF[l1] + jj) * TP_DIM + (IRREP_OFF[l2] + ll)] = (float)acc;
        }
    }
}

// M[c][i][j] = sum_p weights[c][p] * T[p][i][j]   (128 x 16 x 16)
__global__ void fold_weights_kernel(const float* __restrict__ w,
                                    const float* __restrict__ T,
                                    float* __restrict__ M) {
    int c = blockIdx.x;
    int e = threadIdx.x;  // 0..255
    float s = 0.0f;
    #pragma unroll
    for (int p = 0; p < TP_NPATHS; ++p) s += w[c * TP_NPATHS + p] * T[p * 256 + e];
    M[c * 256 + e] = s;
}

// ---------------------------------------------------------------------------
// Multi-value butterfly combine: merges two per-lane partial-sum registers.
// Lanes with (lane & m)==0 end up carrying sum-pairs of `a`, lanes with
// (lane & m)!=0 carry sum-pairs of `b`. Cost: 2 cndmask + 1 bpermute + 1 add.
// ---------------------------------------------------------------------------
__device__ __forceinline__ float combine_xor(float a, float b, int m, int lane) {
    bool hi = (lane & m) != 0;
    float keep = hi ? b : a;
    float send = hi ? a : b;
    return keep + __shfl_xor(send, m, 32);
}

// ---------------------------------------------------------------------------
// Main kernel: per wave, one channel c + a strided set of 16-batch tiles.
//   Z = Y_tile(16x16) x M[c]^T  via 4x V_WMMA_F32_16X16X4_F32
//   out[b] = sum_i X[b][i] * Z[b][i]  via elementwise mult + combine tree
// Row stride between consecutive b (fixed c) = 128*16 = 2048 floats.
// ---------------------------------------------------------------------------
__global__ __launch_bounds__(256) void tp_wmma_kernel(const float* __restrict__ x,
                                                      const float* __restrict__ y,
                                                      const float* __restrict__ M,
                                                      float* __restrict__ out) {
    const int lane = threadIdx.x & 31;
    const int wave = threadIdx.x >> 5;
    const int hh   = lane >> 4;   // lane half (0: lanes 0-15, 1: lanes 16-31)
    const int lrow = lane & 15;
    const int c    = blockIdx.x;  // channel
    const int gwave = blockIdx.y * 8 + wave;
    const int G     = gridDim.y * 8;

    // B fragments of M^T (4x16 each, K split 4x4):
    //   vgpr0 holds K=2*hh, vgpr1 holds K=2*hh+1; N = lrow
    //   B_k[K][N] = M^T[4k+K][N] = M[N][4k+K]  -> adjacent pair -> one b64 load
    const float* Mc = M + (long)c * 256;
    v2f bfrag[4];
    #pragma unroll
    for (int k = 0; k < 4; ++k)
        bfrag[k] = *(const v2f*)(Mc + lrow * TP_DIM + 4 * k + 2 * hh);

    const long rowstride = (long)TP_NCH * TP_DIM;   // 2048 floats per batch row
    const long tilestep  = 16 * rowstride;          // floats per 16-batch tile
    const long step      = (long)G * tilestep;      // loop stride in floats

    // Per-lane stream bases (pointer-bumped each iteration):
    //  A fragments: row b0+lrow, cols {4k+2*hh, 4k+2*hh+1}
    //  X (D layout): row b0 + r + 8*hh, col lrow; r*rowstride folds into offsets
    const float* yp = y + (long)c * TP_DIM + (long)gwave * tilestep
                        + (long)lrow * rowstride + 2 * hh;
    const float* xp = x + (long)c * TP_DIM + (long)gwave * tilestep
                        + (long)(8 * hh) * rowstride + lrow;
    float* op = out + (long)gwave * 16 * TP_NCH + ((long)(lane & 7) + 8 * hh) * TP_NCH + c;
    const long ostep = (long)G * 16 * TP_NCH;
    const bool writer = (lane & 8) == 0;   // lanes 0-7 and 16-23 hold the results

    for (int t = gwave; t < TP_NTILES; t += G, yp += step, xp += step, op += ostep) {
        // A fragments (one b64 per k)
        v2f a[4];
        #pragma unroll
        for (int k = 0; k < 4; ++k)
            a[k] = *(const v2f*)(yp + 4 * k);

        v8f z = {};
        #pragma unroll
        for (int k = 0; k < 4; ++k)
            z = __builtin_amdgcn_wmma_f32_16x16x4_f32(
                    /*neg_a=*/false, a[k], /*neg_b=*/false, bfrag[k],
                    /*c_mod=*/(short)0, z, /*reuse_a=*/false, /*reuse_b=*/false);

        // X in the D layout (coalesced b32 loads, immediate offsets r*8KB)
        float acc[8];
        #pragma unroll
        for (int r = 0; r < 8; ++r)
            acc[r] = xp[(long)r * rowstride] * z[r];

        // branchless prefetch of the next tile (clamped; prefetch never faults)
        {
            long pf = (t + G < TP_NTILES) ? step : 0;
            __builtin_prefetch(yp + pf, 0, 3);
            __builtin_prefetch(xp + pf, 0, 3);
        }

        // combine tree: 8 values x 16 lanes -> 16 sums in 8 DS ops
        float c0 = combine_xor(acc[0], acc[1], 1, lane);
        float c1 = combine_xor(acc[2], acc[3], 1, lane);
        float c2 = combine_xor(acc[4], acc[5], 1, lane);
        float c3 = combine_xor(acc[6], acc[7], 1, lane);
        float d0 = combine_xor(c0, c1, 2, lane);
        float d1 = combine_xor(c2, c3, 2, lane);
        float e  = combine_xor(d0, d1, 4, lane);
        e += __shfl_xor(e, 8, 32);
        // lane l now holds out-row b0 + (l&7) + 8*hh (duplicated at l^8)

        if (writer) *op = e;
    }
}

// ---------------------------------------------------------------------------
// Launch
// ---------------------------------------------------------------------------
extern "C" void kernel_launch(void* const* d_in, const int* in_sizes, int n_in,
                              void* d_out, int out_size, void* d_ws, size_t ws_size,
                              hipStream_t stream) {
    const float* x = (const float*)d_in[0];   // [50000,128,16]
    const float* y = (const float*)d_in[1];   // [50000,128,16]
    const float* w = (const float*)d_in[2];   // [128,23]
    float* out = (float*)d_out;               // [50000,128]

    // workspace layout: T [23*256 floats] then M [128*256 floats] (~155 KB)
    float* T = (float*)d_ws;
    float* M = T + TP_NPATHS * 256;

    build_T_kernel<<<1, 32, 0, stream>>>(T);
    fold_weights_kernel<<<TP_NCH, 256, 0, stream>>>(w, T, M);

    dim3 grid(TP_NCH, 16);   // 128 channels x 16 -> 2048 blocks, 8 waves each
    tp_wmma_kernel<<<grid, 256, 0, stream>>>(x, y, M, out);
}